// EuclideanNet_4964982194203
// MI455X (gfx1250) — compile-verified
//
#include <hip/hip_runtime.h>
#include <math.h>

typedef __attribute__((ext_vector_type(2))) float v2f;
typedef __attribute__((ext_vector_type(8))) float v8f;

#define N_ATOMS 286
#define C_IN    23
#define H_DIM   100
#define H_PAD   112              // 7 tiles of 16
#define B_SZ    32
#define M_ROWS  (B_SZ * N_ATOMS) // 9152 = 572 * 16 exactly
#define NT_G    7
#define MT_G    (M_ROWS / 16)
#define TILES_G (MT_G * NT_G)    // 4004
// Y0 / sqrt(N) = (1/(2*sqrt(pi))) / sqrt(286)
#define SCALE_F (0.28209479177387814f / 16.911534525287763f)

__device__ __forceinline__ v8f wmma_f32_4(v2f a, v2f b, v8f c) {
  // 8 args: (neg_a, A, neg_b, B, c_mod, C, reuse_a, reuse_b)
  return __builtin_amdgcn_wmma_f32_16x16x4_f32(false, a, false, b, (short)0, c,
                                               false, false);
}

// Branchless guarded load: unconditional load from a selected in-bounds index
// (v_cndmask on the address), then value select. Avoids exec-mask juggling.
__device__ __forceinline__ float loadf_pred(const float* __restrict__ p,
                                            int idx, bool pred) {
  float v = p[pred ? idx : 0];
  return pred ? v : 0.0f;
}

__device__ __forceinline__ float basisf(float r, int i) {
  // e3nn cosine basis: radii = [0, 1.5, 3.0], step = 1.5
  float zz = (r - 1.5f * (float)i) * (1.0f / 1.5f);
  float t  = 1.0f - fmaxf(2.0f - fmaxf(zz + 1.0f, 0.0f), 0.0f);
  return __cosf(1.57079632679489662f * t);
}

// ---------------- Kernel 1: G[m, k] = sum_j feat[m, j] * rW2[k, j] ----------
// One wave per 16x16 output tile; 6 chained K=4 WMMAs (K padded 23->24).
__global__ __launch_bounds__(256) void g_kernel(const float* __restrict__ feat,
                                                const float* __restrict__ rW2,
                                                float* __restrict__ G) {
  int wave = blockIdx.x * (blockDim.x >> 5) + (threadIdx.x >> 5);
  int lane = threadIdx.x & 31;
  if (wave >= TILES_G) return;           // wave-uniform branch
  int mt = wave / NT_G, nt = wave % NT_G;
  int hi = lane >> 4;                    // 0: K rows {0,1}; 1: K rows {2,3}
  int m  = mt * 16 + (lane & 15);
  int kh = nt * 16 + (lane & 15);
  bool kv = kh < H_DIM;
  const float* fr = feat + (size_t)m * C_IN;
  const float* wr = rW2 + (size_t)(kv ? kh : 0) * C_IN;

  v8f acc = {};
#pragma unroll
  for (int ks = 0; ks < 6; ++ks) {
    int j0 = ks * 4 + hi * 2;
    v2f a, b;
    a.x = loadf_pred(fr, j0,     j0     < C_IN);
    a.y = loadf_pred(fr, j0 + 1, j0 + 1 < C_IN);
    b.x = loadf_pred(wr, j0,     kv && (j0     < C_IN));
    b.y = loadf_pred(wr, j0 + 1, kv && (j0 + 1 < C_IN));
    acc = wmma_f32_4(a, b, acc);
  }
#pragma unroll
  for (int i = 0; i < 8; ++i) {
    int row = mt * 16 + hi * 8 + i;      // D row = vgpr + 8*(lane>=16)
    G[(size_t)row * H_PAD + nt * 16 + (lane & 15)] = acc[i];
  }
}

// ---------------- Kernel 2: fused conv --------------------------------------
// block = (z, a-tile); 8 waves split the b-loop; per b: basis -> 7 WMMAs ->
// relu -> FMA vs G[b,:]; k-contraction lives in registers the whole time.
__global__ __launch_bounds__(256) void conv_kernel(const float* __restrict__ geom,
                                                   const float* __restrict__ rW1,
                                                   const float* __restrict__ rb1,
                                                   const float* __restrict__ G,
                                                   float* __restrict__ feats) {
  __shared__ float sgeom[N_ATOMS * 3];
  __shared__ float sacc[16];
  int z = blockIdx.x, at = blockIdx.y;
  int tid = threadIdx.x, lane = tid & 31, w = tid >> 5;

  for (int i = tid; i < N_ATOMS * 3; i += 256) sgeom[i] = geom[z * N_ATOMS * 3 + i];
  if (tid < 16) sacc[tid] = 0.0f;
  __syncthreads();

  int hi = lane >> 4;
  int a  = at * 16 + (lane & 15);
  int ac = (a < N_ATOMS) ? a : (N_ATOMS - 1);   // clamp; garbage rows discarded
  float ax = sgeom[ac * 3 + 0], ay = sgeom[ac * 3 + 1], az = sgeom[ac * 3 + 2];

  // Resident B tiles (rW1, zero-padded K row 3 and cols >= 100) + biases.
  v2f Bt[7];
  float bias[7];
#pragma unroll
  for (int t = 0; t < 7; ++t) {
    int kc = t * 16 + (lane & 15);
    bool v = kc < H_DIM;
    int r0 = hi * 2;
    Bt[t].x = loadf_pred(rW1, r0 * H_DIM + kc,       v);            // rows 0 / 2
    Bt[t].y = loadf_pred(rW1, (r0 + 1) * H_DIM + kc, v && !hi);     // rows 1 / pad
    bias[t] = loadf_pred(rb1, kc, v);
  }

  v8f acc = {};
  const float* Gz = G + (size_t)z * N_ATOMS * H_PAD;
#pragma unroll 2
  for (int b = w; b < N_ATOMS; b += 8) {       // wave-uniform trip count
    float bx = sgeom[b * 3 + 0], by = sgeom[b * 3 + 1], bz = sgeom[b * 3 + 2];
    float dx = ax - bx, dy = ay - by, dz = az - bz;
    float r = sqrtf(fmaf(dx, dx, fmaf(dy, dy, fmaf(dz, dz, 1e-12f))));
    v2f A;
    A.x = basisf(r, hi * 2);                   // basis rows {0,1} / {2, pad}
    A.y = hi ? 0.0f : basisf(r, 1);
    const float* gb = Gz + (size_t)b * H_PAD + (lane & 15);
    v8f dzero = {};
#pragma unroll
    for (int t = 0; t < 7; ++t) {
      v8f d = wmma_f32_4(A, Bt[t], dzero);
      float g = gb[t * 16];                    // G[b, k] for this lane's column
#pragma unroll
      for (int i = 0; i < 8; ++i) {
        float h = fmaxf(d[i] + bias[t], 0.0f); // relu(basis@rW1 + rb1)
        acc[i] = fmaf(h, g, acc[i]);
      }
    }
  }

  // Reduce the k-dimension across the 16 lanes of each half (same D rows).
#pragma unroll
  for (int off = 1; off <= 8; off <<= 1) {
#pragma unroll
    for (int i = 0; i < 8; ++i) acc[i] += __shfl_xor(acc[i], off, 32);
  }
  if ((lane & 15) == 0) {
#pragma unroll
    for (int i = 0; i < 8; ++i) atomicAdd(&sacc[hi * 8 + i], acc[i]);
  }
  __syncthreads();
  if (tid < 16) {
    int aa = at * 16 + tid;
    if (aa < N_ATOMS) feats[z * N_ATOMS + aa] = sacc[tid] * SCALE_F;
  }
}

// ---------------- Kernel 3: rb2 bias term + MLP head ------------------------
__global__ __launch_bounds__(32) void head_kernel(
    const float* __restrict__ feat, const float* __restrict__ rb2,
    const float* __restrict__ feats, const float* __restrict__ fc1W,
    const float* __restrict__ fc1b, const float* __restrict__ fc2W,
    const float* __restrict__ fc2b, const float* __restrict__ fc3W,
    const float* __restrict__ fc3b, float* __restrict__ out) {
  __shared__ float t1[30], t2[10];
  int z = blockIdx.x, tid = threadIdx.x;
  float s = 0.0f;  // sum_b sum_j rb2[j]*feat[z,b,j]
  for (int i = tid; i < N_ATOMS * C_IN; i += 32)
    s += feat[z * N_ATOMS * C_IN + i] * rb2[i % C_IN];
  for (int off = 1; off < 32; off <<= 1) s += __shfl_xor(s, off, 32);
  float biasS = s * SCALE_F;

  if (tid < 30) {
    float acc = fc1b[tid];
    for (int a2 = 0; a2 < N_ATOMS; ++a2)
      acc = fmaf(feats[z * N_ATOMS + a2] + biasS, fc1W[a2 * 30 + tid], acc);
    t1[tid] = fmaxf(acc, 0.0f);
  }
  __syncthreads();
  if (tid < 10) {
    float acc = fc2b[tid];
    for (int o = 0; o < 30; ++o) acc = fmaf(t1[o], fc2W[o * 10 + tid], acc);
    t2[tid] = fmaxf(acc, 0.0f);
  }
  __syncthreads();
  if (tid == 0) {
    float acc = fc3b[0];
    for (int p = 0; p < 10; ++p) acc = fmaf(t2[p], fc3W[p], acc);
    out[z] = acc;
  }
}

extern "C" void kernel_launch(void* const* d_in, const int* in_sizes, int n_in,
                              void* d_out, int out_size, void* d_ws, size_t ws_size,
                              hipStream_t stream) {
  // setup_inputs order: x, features, geometry, rW1, rb1, rW2, rb2,
  //                     fc1W, fc1b, fc2W, fc2b, fc3W, fc3b   (x unused)
  const float* features = (const float*)d_in[1];
  const float* geometry = (const float*)d_in[2];
  const float* rW1  = (const float*)d_in[3];
  const float* rb1  = (const float*)d_in[4];
  const float* rW2  = (const float*)d_in[5];
  const float* rb2  = (const float*)d_in[6];
  const float* fc1W = (const float*)d_in[7];
  const float* fc1b = (const float*)d_in[8];
  const float* fc2W = (const float*)d_in[9];
  const float* fc2b = (const float*)d_in[10];
  const float* fc3W = (const float*)d_in[11];
  const float* fc3b = (const float*)d_in[12];
  float* out = (float*)d_out;

  // workspace: G [9152 x 112] f32 (4.1 MB) + feats [32 x 286] f32
  float* G     = (float*)d_ws;
  float* feats = G + (size_t)M_ROWS * H_PAD;

  int blocks1 = (TILES_G + 7) / 8;   // 8 waves/block, one tile per wave
  g_kernel<<<blocks1, 256, 0, stream>>>(features, rW2, G);

  dim3 grid2(B_SZ, (N_ATOMS + 15) / 16);  // 32 x 18
  conv_kernel<<<grid2, 256, 0, stream>>>(geometry, rW1, rb1, G, feats);

  head_kernel<<<B_SZ, 32, 0, stream>>>(features, rb2, feats, fc1W, fc1b, fc2W,
                                       fc2b, fc3W, fc3b, out);
}